// Render_Land_32091995635815
// MI455X (gfx1250) — compile-verified
//
#include <hip/hip_runtime.h>
#include <math.h>

typedef __attribute__((ext_vector_type(2))) float v2f;
typedef __attribute__((ext_vector_type(8))) float v8f;

#define INF_BITS 0x7F800000u

// ---------------------------------------------------------------- init
__global__ void k_init(unsigned* dmin, float* acc, int n) {
  int i = blockIdx.x * blockDim.x + threadIdx.x;
  if (i < n) dmin[i] = INF_BITS;
  if (i < 4) acc[i] = 0.0f;
}

// ---------------------------------------------------------------- euler -> R (Rz@Ry@Rx)
__global__ void k_rot(const float* euler, float* R, int B) {
  int b = threadIdx.x;
  if (b >= B) return;
  float x = euler[b * 3 + 0], y = euler[b * 3 + 1], z = euler[b * 3 + 2];
  float cx = cosf(x), sx = sinf(x);
  float cy = cosf(y), sy = sinf(y);
  float cz = cosf(z), sz = sinf(z);
  // Rzy = Rz @ Ry
  float zy[9] = { cz * cy, -sz, cz * sy,
                  sz * cy,  cz, sz * sy,
                  -sy,     0.f, cy };
  float* r = R + b * 9;
  for (int i = 0; i < 3; ++i) {
    r[i * 3 + 0] = zy[i * 3 + 0];
    r[i * 3 + 1] = zy[i * 3 + 1] * cx + zy[i * 3 + 2] * sx;
    r[i * 3 + 2] = -zy[i * 3 + 1] * sx + zy[i * 3 + 2] * cx;
  }
}

// ---------------------------------------------------------------- vertex transform + projection + seg
__global__ void k_vert1(const float* geom, const float* R, const float* trans,
                        const float* cam, float* rott, float* proj, int* seg,
                        int B, int N, int W, int H) {
  int i = blockIdx.x * blockDim.x + threadIdx.x;
  if (i >= B * N) return;
  int b = i / N;
  float gx = geom[i * 3 + 0], gy = geom[i * 3 + 1], gz = geom[i * 3 + 2];
  const float* r = R + b * 9;
  float X = r[0] * gx + r[1] * gy + r[2] * gz + trans[b * 3 + 0];
  float Y = r[3] * gx + r[4] * gy + r[5] * gz + trans[b * 3 + 1];
  float Z = r[6] * gx + r[7] * gy + r[8] * gz + trans[b * 3 + 2];
  rott[i * 3 + 0] = X; rott[i * 3 + 1] = Y; rott[i * 3 + 2] = Z;
  float fx = cam[b * 4 + 0], fy = cam[b * 4 + 1];
  float cxx = cam[b * 4 + 2], cyy = cam[b * 4 + 3];
  float u = fx * X / Z + cxx;
  float v = fy * Y / Z + cyy;
  proj[i * 3 + 0] = u; proj[i * 3 + 1] = v; proj[i * 3 + 2] = Z;
  float pu = fminf(fmaxf(rintf(u), 0.f), (float)(W - 1));
  float pv = fminf(fmaxf(rintf(v), 0.f), (float)(H - 1));
  seg[i] = b * (H * W) + (int)pv * W + (int)pu;
}

// ---------------------------------------------------------------- triangle normals + SH9, nbl = sh @ light via WMMA f32 16x16x4
__global__ void __launch_bounds__(256) k_tri(const float* rott, const int* tris,
                                             const float* light, float* tri_n,
                                             float* nbl, int B, int N, int T) {
  __shared__ float shl[8][32][12];  // per-wave 32 triangles x K=12 (SH9 zero-padded)
  int b = blockIdx.y;
  int tid = threadIdx.x;
  int wave = tid >> 5, lane = tid & 31;
  int tri = blockIdx.x * 256 + tid;
  int t = tri < T ? tri : T - 1;  // clamp tail; keep EXEC uniform for WMMA

  int i0 = tris[t * 3 + 0], i1 = tris[t * 3 + 1], i2 = tris[t * 3 + 2];
  const float* p0 = rott + ((size_t)b * N + i0) * 3;
  const float* p1 = rott + ((size_t)b * N + i1) * 3;
  const float* p2 = rott + ((size_t)b * N + i2) * 3;
  float e1x = p1[0] - p0[0], e1y = p1[1] - p0[1], e1z = p1[2] - p0[2];
  float e2x = p2[0] - p0[0], e2y = p2[1] - p0[1], e2z = p2[2] - p0[2];
  float nx = e1y * e2z - e1z * e2y;
  float ny = e1z * e2x - e1x * e2z;
  float nz = e1x * e2y - e1y * e2x;
  float inv = 1.0f / sqrtf(nx * nx + ny * ny + nz * nz + 1e-12f);
  nx *= inv; ny *= inv; nz *= inv;
  if (tri < T) {
    float* tn = tri_n + ((size_t)b * T + tri) * 3;
    tn[0] = nx; tn[1] = ny; tn[2] = nz;
  }
  // SH9 row into LDS (K padded to 12)
  const float K0 = 0.88622689f;   // a0*c0
  const float K1 = 1.77245385f;   // a1*c1
  const float K2 = 2.4270337f;    // a2*c2
  float* s = shl[wave][lane];
  s[0] = K0;
  s[1] = -K1 * ny; s[2] = K1 * nz; s[3] = -K1 * nx;
  s[4] = K2 * nx * ny; s[5] = -K2 * ny * nz;
  s[6] = K2 * 0.28867513f * (3.f * nz * nz - 1.f);
  s[7] = -K2 * nx * nz; s[8] = K2 * 0.5f * (nx * nx - ny * ny);
  s[9] = 0.f; s[10] = 0.f; s[11] = 0.f;
  __syncthreads();

  int laneN = lane & 15, hi = lane >> 4;
  // B fragments: light reshaped 9x3, zero-padded to 12x16.
  // 4x16 f32 B layout: V0 = K rows {k0, k0+2} split across lane halves, V1 = {k0+1, k0+3}.
  v2f bf[3];
  const float* Lp = light + b * 27;
#pragma unroll
  for (int sgp = 0; sgp < 3; ++sgp) {
    int k0 = sgp * 4 + hi * 2;
    bf[sgp].x = (k0 < 9 && laneN < 3) ? Lp[k0 * 3 + laneN] : 0.f;
    bf[sgp].y = (k0 + 1 < 9 && laneN < 3) ? Lp[(k0 + 1) * 3 + laneN] : 0.f;
  }
  v8f d0 = {}; v8f d1 = {};
#pragma unroll
  for (int sgp = 0; sgp < 3; ++sgp) {
    int k0 = sgp * 4 + hi * 2;
    v2f a0, a1;  // 16x4 f32 A layout: lane M=laneN, lane-half selects K pair
    a0.x = shl[wave][laneN][k0];        a0.y = shl[wave][laneN][k0 + 1];
    a1.x = shl[wave][16 + laneN][k0];   a1.y = shl[wave][16 + laneN][k0 + 1];
    d0 = __builtin_amdgcn_wmma_f32_16x16x4_f32(false, a0, false, bf[sgp],
                                               (short)0, d0, false, false);
    d1 = __builtin_amdgcn_wmma_f32_16x16x4_f32(false, a1, false, bf[sgp],
                                               (short)0, d1, false, false);
  }
  // Stage D tiles through the (now dead) LDS slab so each lane can emit one
  // contiguous, coalesced 12B store instead of 16 exec-masked scalar stores.
  // D layout: VGPR r -> row M = r + 8*hi, col N = laneN; only N<3 are real.
  if (laneN < 3) {
#pragma unroll
    for (int r = 0; r < 8; ++r) {
      int m = r + hi * 8;
      shl[wave][m][laneN]      = d0[r];
      shl[wave][16 + m][laneN] = d1[r];
    }
  }
  __syncthreads();
  if (tri < T) {
    float* np = nbl + ((size_t)b * T + tri) * 3;
    np[0] = shl[wave][lane][0];
    np[1] = shl[wave][lane][1];
    np[2] = shl[wave][lane][2];
  }
}

// ---------------------------------------------------------------- per-vertex vis/color/depth + segment-min
__global__ void k_vert2(const float* rott, const float* proj, const int* vert_tris,
                        const float* tri_n, const float* nbl, const float* texture,
                        const int* seg, float* color, float* depth, unsigned* dmin,
                        int B, int N, int T) {
  int i = blockIdx.x * blockDim.x + threadIdx.x;
  if (i >= B * N) return;
  int b = i / N, n = i - b * N;
  int t = vert_tris[n];
  size_t bt = ((size_t)b * T + t) * 3;
  float vnx = tri_n[bt + 0], vny = tri_n[bt + 1], vnz = tri_n[bt + 2];
  float rx = rott[i * 3 + 0], ry = rott[i * 3 + 1], rz = rott[i * 3 + 2];
  float invl = 1.0f / sqrtf(rx * rx + ry * ry + rz * rz + 1e-12f);
  float vis = -(vnx * rx + vny * ry + vnz * rz) * invl;
  if (vis < 0.01f) vis = -1.0f;
  color[i * 3 + 0] = texture[i * 3 + 0] * nbl[bt + 0];
  color[i * 3 + 1] = texture[i * 3 + 1] * nbl[bt + 1];
  color[i * 3 + 2] = texture[i * 3 + 2] * nbl[bt + 2];
  float Z = proj[i * 3 + 2];
  float d = (vis > 0.f) ? Z : 1e9f;
  depth[i] = d;
  // Z > 0 here, so uint-bit compare == float compare
  if (vis > 0.f) atomicMin(&dmin[seg[i]], __float_as_uint(Z));
}

// ---------------------------------------------------------------- winner scatter
__global__ void k_vert3(const float* color, const float* depth, const int* seg,
                        const unsigned* dmin, float* render, int BN) {
  int i = blockIdx.x * blockDim.x + threadIdx.x;
  if (i >= BN) return;
  float d = depth[i];
  if (d >= 1e9f) return;  // invisible
  int sgi = seg[i];
  if (__float_as_uint(d) <= dmin[sgi]) {
    render[(size_t)sgi * 3 + 0] = color[i * 3 + 0];
    render[(size_t)sgi * 3 + 1] = color[i * 3 + 1];
    render[(size_t)sgi * 3 + 2] = color[i * 3 + 2];
  }
}

// ---------------------------------------------------------------- pixel loss reduction
__global__ void __launch_bounds__(256) k_pix(const float* render, const float* real,
                                             const unsigned* dmin, float* acc, int BHW) {
  __shared__ float s1[256];
  __shared__ float s2[256];
  int i = blockIdx.x * 256 + threadIdx.x;
  float cm = 0.f, va = 0.f;
  if (i < BHW && dmin[i] != INF_BITS) {  // pixel_valid == (dmin finite)
    float d0 = render[(size_t)i * 3 + 0] - real[(size_t)i * 3 + 0];
    float d1 = render[(size_t)i * 3 + 1] - real[(size_t)i * 3 + 1];
    float d2 = render[(size_t)i * 3 + 2] - real[(size_t)i * 3 + 2];
    cm = sqrtf(d0 * d0 + d1 * d1 + d2 * d2 + 1e-12f);
    va = 1.f;
  }
  s1[threadIdx.x] = cm; s2[threadIdx.x] = va;
  __syncthreads();
  for (int off = 128; off > 0; off >>= 1) {
    if (threadIdx.x < off) {
      s1[threadIdx.x] += s1[threadIdx.x + off];
      s2[threadIdx.x] += s2[threadIdx.x + off];
    }
    __syncthreads();
  }
  if (threadIdx.x == 0) {
    atomicAdd(acc + 0, s1[0]);
    atomicAdd(acc + 1, s2[0]);
  }
}

// ---------------------------------------------------------------- landmark loss
__global__ void k_land(const float* proj, const int* lands_info, const float* lands,
                       float* acc, int B, int N, int L) {
  int i = blockIdx.x * blockDim.x + threadIdx.x;
  if (i >= B * L) return;
  int b = i / L, l = i - b * L;
  int v = lands_info[l];
  float u = proj[((size_t)b * N + v) * 3 + 0];
  float w = proj[((size_t)b * N + v) * 3 + 1];
  float du = u - lands[i * 2 + 0];
  float dv = w - lands[i * 2 + 1];
  atomicAdd(acc + 2, sqrtf(du * du + dv * dv + 1e-12f));
}

// ---------------------------------------------------------------- finalize 2 scalars
__global__ void k_final(const float* acc, float* out, int BHW, int BL) {
  if (threadIdx.x == 0 && blockIdx.x == 0) {
    float m1 = acc[0] / (float)BHW;
    float m2 = acc[1] / (float)BHW;
    out[0] = m1 / (m2 + 1e-5f);
    out[1] = acc[2] / (float)BL;
  }
}

extern "C" void kernel_launch(void* const* d_in, const int* in_sizes, int n_in,
                              void* d_out, int out_size, void* d_ws, size_t ws_size,
                              hipStream_t stream) {
  const float* geom      = (const float*)d_in[0];
  const float* euler     = (const float*)d_in[1];
  const float* trans     = (const float*)d_in[2];
  const float* cam       = (const float*)d_in[3];
  const float* ori       = (const float*)d_in[4];
  const float* light     = (const float*)d_in[5];
  const float* tex       = (const float*)d_in[6];
  const float* lands     = (const float*)d_in[7];
  const int*   tris      = (const int*)d_in[8];
  const int*   vert_tris = (const int*)d_in[9];
  const int*   lands_inf = (const int*)d_in[10];

  int B  = in_sizes[1] / 3;
  int N  = in_sizes[9];
  int T  = in_sizes[8] / 3;
  int L  = in_sizes[10];
  int HW = in_sizes[4] / (3 * B);
  int W  = 1;
  while ((long long)(W + 1) * (W + 1) <= (long long)HW) ++W;
  int H = HW / W;
  int BN = B * N, BHW = B * HW, BL = B * L;

  // workspace carve-up (all float/int/uint, 4B aligned)
  float*    R      = (float*)d_ws;
  float*    rott   = R + (size_t)B * 9;
  float*    proj   = rott + (size_t)BN * 3;
  int*      seg    = (int*)(proj + (size_t)BN * 3);
  float*    depth  = (float*)(seg + BN);
  float*    color  = depth + BN;
  float*    tri_n  = color + (size_t)BN * 3;
  float*    nbl    = tri_n + (size_t)B * T * 3;
  unsigned* dmin   = (unsigned*)(nbl + (size_t)B * T * 3);
  float*    render = (float*)(dmin + BHW);
  float*    acc    = render + (size_t)BHW * 3;

  float* out = (float*)d_out;

  int g_init = (BHW + 255) / 256;
  k_init<<<g_init, 256, 0, stream>>>(dmin, acc, BHW);

  k_rot<<<1, 64, 0, stream>>>(euler, R, B);

  int g_v = (BN + 255) / 256;
  k_vert1<<<g_v, 256, 0, stream>>>(geom, R, trans, cam, rott, proj, seg, B, N, W, H);

  dim3 g_t((T + 255) / 256, B);
  k_tri<<<g_t, 256, 0, stream>>>(rott, tris, light, tri_n, nbl, B, N, T);

  k_vert2<<<g_v, 256, 0, stream>>>(rott, proj, vert_tris, tri_n, nbl, tex, seg,
                                   color, depth, dmin, B, N, T);

  k_vert3<<<g_v, 256, 0, stream>>>(color, depth, seg, dmin, render, BN);

  k_pix<<<g_init, 256, 0, stream>>>(render, ori, dmin, acc, BHW);

  k_land<<<(BL + 63) / 64, 64, 0, stream>>>(proj, lands_inf, lands, acc, B, N, L);

  k_final<<<1, 32, 0, stream>>>(acc, out, BHW, BL);
}